// Allinone_plus_30657476559424
// MI455X (gfx1250) — compile-verified
//
#include <hip/hip_runtime.h>
#include <hip/hip_bf16.h>

// Problem constants (match reference).
#define N_NODES 100000
#define T_DIM   64
#define D_DIM   128
#define M_MATS  8

typedef __attribute__((ext_vector_type(2))) float v2f;
typedef __attribute__((ext_vector_type(4))) float v4f;
typedef __attribute__((ext_vector_type(8))) float v8f;

// ---------------------------------------------------------------------------
// Kernel 1: fold head weights into the cross mats once:
//   w_mat[t,d] = sum_m head_w[m] * mats[m,t,d]      (64x128 -> d_ws)
// Tiny; avoids every GEMM block re-reading the full 256KB mats tensor.
// ---------------------------------------------------------------------------
__global__ __launch_bounds__(256)
void fold_head_kernel(const float* __restrict__ mats,
                      const float* __restrict__ head_w,
                      float* __restrict__ w_mat) {
  const int i = blockIdx.x * 256 + threadIdx.x;   // 0 .. 8191
  if (i < T_DIM * D_DIM) {
    float acc = 0.0f;
#pragma unroll
    for (int m = 0; m < M_MATS; ++m)
      acc = fmaf(head_w[m], mats[m * (T_DIM * D_DIM) + i], acc);
    w_mat[i] = acc;
  }
}

// ---------------------------------------------------------------------------
// Kernel 2: out[t,n] = prune(sigmoid(w_mat[t,:] . x[n,:] + b))
// via V_WMMA_F32_16X16X4_F32. Block = 256 threads = 8 wave32s; each wave
// computes a 64(T) x 16(N) strip -> block covers 128 columns of N.
// ---------------------------------------------------------------------------
#define LDS_STRIDE 132   // 128 + 4 floats pad: A-fragment ds_load_b64 reads hit
                         // banks (4*lrow + 2*half) -> all 32 lanes distinct.

__global__ __launch_bounds__(256)
void cross_gemm_kernel(const float* __restrict__ x,
                       const float* __restrict__ w_mat,
                       const float* __restrict__ head_b,
                       float* __restrict__ out) {
  __shared__ float lds_w[T_DIM * LDS_STRIDE];     // 33,792 B

  // Cooperative, coalesced load of w_mat into padded LDS (float4 chunks).
  for (int j = threadIdx.x; j < (T_DIM * D_DIM) / 4; j += 256) {
    const int row = (j * 4) / D_DIM;
    const int col = (j * 4) % D_DIM;
    *(v4f*)(&lds_w[row * LDS_STRIDE + col]) = *(const v4f*)(w_mat + j * 4);
  }
  __syncthreads();

  const float hb   = head_b[0];
  const int   lane = threadIdx.x & 31;   // wave32
  const int   wave = threadIdx.x >> 5;   // 0..7
  const int   half = lane >> 4;          // upper half-wave selects K pair / M+8
  const int   lrow = lane & 15;

  // This lane's N column (x row). Clamp for the tail: WMMA needs EXEC==all 1s,
  // so out-of-range lanes compute on a valid row and are masked at the store.
  const int n   = blockIdx.x * 128 + wave * 16 + lrow;
  const int n_c = (n < N_NODES) ? n : (N_NODES - 1);
  const float* __restrict__ xrow = x + (size_t)n_c * D_DIM;

  // 4 accumulator tiles covering T = 0..63.
  v8f acc[4] = {v8f{}, v8f{}, v8f{}, v8f{}};

  // K loop: D = 128 = 32 steps of K=4.
#pragma unroll
  for (int ks = 0; ks < D_DIM / 4; ++ks) {
    const int kofs = ks * 4 + half * 2;          // ISA A/B f32 16x16x4 layout
    const v2f b = *(const v2f*)(xrow + kofs);    // B: x^T fragment (global)
#pragma unroll
    for (int tt = 0; tt < 4; ++tt) {
      const v2f a = *(const v2f*)(&lds_w[(tt * 16 + lrow) * LDS_STRIDE + kofs]);
      // (neg_a, A, neg_b, B, c_mod, C, reuse_a, reuse_b)
      acc[tt] = __builtin_amdgcn_wmma_f32_16x16x4_f32(
          false, a, false, b, (short)0, acc[tt], false, false);
    }
  }

  // Epilogue: sigmoid + prune. sigmoid(v) < 0.1  <=>  v < ln(1/9).
  if (n < N_NODES) {
    const float thresh = -2.1972245773362196f;   // logit(0.1)
#pragma unroll
    for (int tt = 0; tt < 4; ++tt) {
#pragma unroll
      for (int r = 0; r < 8; ++r) {
        const int   t = tt * 16 + half * 8 + r;  // C/D layout: VGPR r, halfwave
        const float v = acc[tt][r] + hb;
        const float s = (v < thresh) ? 0.0f : (1.0f / (1.0f + __expf(-v)));
        out[(size_t)t * N_NODES + n] = s;
      }
    }
  }
}

// ---------------------------------------------------------------------------
extern "C" void kernel_launch(void* const* d_in, const int* in_sizes, int n_in,
                              void* d_out, int out_size, void* d_ws, size_t ws_size,
                              hipStream_t stream) {
  const float* x      = (const float*)d_in[0];   // [N, D]
  const float* mats   = (const float*)d_in[1];   // [M, T, D]
  const float* head_w = (const float*)d_in[2];   // [M]
  const float* head_b = (const float*)d_in[3];   // scalar
  float*       out    = (float*)d_out;           // [T, N]
  float*       w_mat  = (float*)d_ws;            // [T, D] scratch (32 KB)

  // Kernel 1: fold head into mats -> w_mat.
  fold_head_kernel<<<(T_DIM * D_DIM + 255) / 256, 256, 0, stream>>>(mats, head_w, w_mat);

  // Kernel 2: WMMA GEMM + sigmoid/prune epilogue. 128 N-columns per block.
  const int nblocks = (N_NODES + 127) / 128;     // 782
  cross_gemm_kernel<<<nblocks, 256, 0, stream>>>(x, w_mat, head_b, out);
}